// KTop_61237643706890
// MI455X (gfx1250) — compile-verified
//
#include <hip/hip_runtime.h>
#include <stdint.h>

// Problem geometry (from reference): z is 4096 x 32768 f32, k = 64.
#define NCOLS  32768
#define TPB    1024
#define CHUNKS 8                      // (NCOLS/4) / TPB float4 chunks per thread
#define ROW_BYTES (NCOLS * 4)
// dynamic LDS: async landing buffer | hist[256] | scanA[256] | scanB[256] | scalars
#define SMEM_BYTES (ROW_BYTES + 256*4 + 256*4 + 256*4 + 16)

#if __has_builtin(__builtin_amdgcn_global_load_async_to_lds_b128)
#define USE_ASYNC_LDS 1
#else
#define USE_ASYNC_LDS 0
#endif

// int4 vectors in explicit address spaces (arg0 = global, arg1 = LDS).
typedef int v4i __attribute__((ext_vector_type(4)));
typedef __attribute__((address_space(1))) v4i gv4i;
typedef __attribute__((address_space(3))) v4i lv4i;

#if __has_builtin(__builtin_amdgcn_s_wait_asynccnt)
#define WAIT_ASYNC(n) do { __builtin_amdgcn_s_wait_asynccnt(n); \
                           asm volatile("" ::: "memory"); } while (0)
#else
#define WAIT_ASYNC(n) asm volatile("s_wait_asynccnt %0" :: "i"(n) : "memory")
#endif

// Order-preserving map: float -> uint32 (larger float <=> larger key).
__device__ __forceinline__ uint32_t f2key(float f) {
    uint32_t b = __float_as_uint(f);
    return (b & 0x80000000u) ? ~b : (b | 0x80000000u);
}

__global__ __launch_bounds__(TPB)
void topk_thresh_kernel(const float* __restrict__ z,
                        const int*   __restrict__ kptr,
                        float*       __restrict__ out) {
    extern __shared__ char dyn_smem[];
    float4*   rowF4 = (float4*)dyn_smem;                   // async landing buffer
    uint32_t* hist  = (uint32_t*)(dyn_smem + ROW_BYTES);
    uint32_t* sA    = hist + 256;
    uint32_t* sB    = sA + 256;
    uint32_t* scal  = sB + 256;    // [0]=prefix, [1]=remaining

    const int tid = threadIdx.x;
    const size_t rowbase = (size_t)blockIdx.x * NCOLS;
    const float* src = z + rowbase;
    float*       dst = out + rowbase;

    const int k_in = kptr[0];
    const bool pass_all = (k_in >= NCOLS);
    uint32_t remaining = (uint32_t)(k_in < 1 ? 1 : (k_in > NCOLS ? NCOLS : k_in));

    if (tid < 256) hist[tid] = 0;

    float4 vals[CHUNKS];   // per-thread register cache of its 32 elements

#if USE_ASYNC_LDS
    // Issue ALL async loads up front: no VGPR destinations, 8 in flight per
    // thread, tracked by the per-wave in-order ASYNCcnt.
#pragma unroll
    for (int j = 0; j < CHUNKS; ++j) {
        const int f4 = tid + j * TPB;
        gv4i* gp = (gv4i*)(uintptr_t)(src + (size_t)f4 * 4);
        lv4i* lp = (lv4i*)(uint32_t)(uintptr_t)(rowF4 + f4);
        __builtin_amdgcn_global_load_async_to_lds_b128(gp, lp, 0, 0);
    }
    __syncthreads();   // hist zeroed before any atomics; loads already in flight

    // Round-0 histogram pipelined against async completion: chunk j of this
    // wave is guaranteed landed once ASYNCcnt <= (7-j).
#define CONSUME(J, CNT)                                            \
    do {                                                           \
        WAIT_ASYNC(CNT);                                           \
        vals[J] = rowF4[tid + (J) * TPB];                          \
        atomicAdd(&hist[f2key(vals[J].x) >> 24], 1u);              \
        atomicAdd(&hist[f2key(vals[J].y) >> 24], 1u);              \
        atomicAdd(&hist[f2key(vals[J].z) >> 24], 1u);              \
        atomicAdd(&hist[f2key(vals[J].w) >> 24], 1u);              \
    } while (0)
    CONSUME(0, 7); CONSUME(1, 6); CONSUME(2, 5); CONSUME(3, 4);
    CONSUME(4, 3); CONSUME(5, 2); CONSUME(6, 1); CONSUME(7, 0);
#undef CONSUME
#else
    __syncthreads();
#pragma unroll
    for (int j = 0; j < CHUNKS; ++j) {
        vals[j] = ((const float4*)src)[tid + j * TPB];
        atomicAdd(&hist[f2key(vals[j].x) >> 24], 1u);
        atomicAdd(&hist[f2key(vals[j].y) >> 24], 1u);
        atomicAdd(&hist[f2key(vals[j].z) >> 24], 1u);
        atomicAdd(&hist[f2key(vals[j].w) >> 24], 1u);
    }
#endif
    __syncthreads();

    // ---- Exact k-th-largest via 4-round radix select (8 bits / round) ----
    uint32_t prefix = 0;
    for (int round = 0; round < 4; ++round) {
        const int shift = 24 - 8 * round;

        if (round > 0) {
            if (tid < 256) hist[tid] = 0;
            __syncthreads();
            const uint32_t himask = 0xFFFFFFFFu << (shift + 8);
#pragma unroll
            for (int j = 0; j < CHUNKS; ++j) {
                float fv[4] = { vals[j].x, vals[j].y, vals[j].z, vals[j].w };
#pragma unroll
                for (int c = 0; c < 4; ++c) {
                    uint32_t key = f2key(fv[c]);
                    if ((key & himask) == prefix)
                        atomicAdd(&hist[(key >> shift) & 0xFFu], 1u);
                }
            }
            __syncthreads();
        }

        // suffix scan: cum[t] = sum_{j>=t} hist[j]  (Hillis-Steele, ping-pong)
        if (tid < 256) sA[tid] = hist[tid];
        __syncthreads();
        uint32_t* cur = sA;
        uint32_t* nxt = sB;
        for (int d = 1; d < 256; d <<= 1) {
            if (tid < 256) {
                uint32_t x = cur[tid];
                if (tid + d < 256) x += cur[tid + d];
                nxt[tid] = x;
            }
            __syncthreads();
            uint32_t* t = cur; cur = nxt; nxt = t;
        }
        // exactly one thread satisfies the crossing condition
        if (tid < 256) {
            uint32_t c  = cur[tid];
            uint32_t cn = (tid < 255) ? cur[tid + 1] : 0u;
            if (c >= remaining && cn < remaining) {
                scal[0] = prefix | ((uint32_t)tid << shift);
                scal[1] = remaining - cn;
            }
        }
        __syncthreads();
        prefix    = scal[0];
        remaining = scal[1];
        __syncthreads();
    }
    const uint32_t kth_key = prefix;   // key of the exact k-th largest value

    // ---- Threshold from registers + single vectorized HBM write ----
#pragma unroll
    for (int j = 0; j < CHUNKS; ++j) {
        float4 v = vals[j];
        float4 o;
        o.x = (pass_all || f2key(v.x) >= kth_key) ? v.x : 0.0f;
        o.y = (pass_all || f2key(v.y) >= kth_key) ? v.y : 0.0f;
        o.z = (pass_all || f2key(v.z) >= kth_key) ? v.z : 0.0f;
        o.w = (pass_all || f2key(v.w) >= kth_key) ? v.w : 0.0f;
        ((float4*)dst)[tid + j * TPB] = o;
    }
}

extern "C" void kernel_launch(void* const* d_in, const int* in_sizes, int n_in,
                              void* d_out, int out_size, void* d_ws, size_t ws_size,
                              hipStream_t stream) {
    const float* z = (const float*)d_in[0];
    const int*   k = (const int*)d_in[1];
    float* out = (float*)d_out;

    const int nrows = out_size / NCOLS;   // 4096
    topk_thresh_kernel<<<nrows, TPB, SMEM_BYTES, stream>>>(z, k, out);
}